// GSWorker_34892314312746
// MI455X (gfx1250) — compile-verified
//
#include <hip/hip_runtime.h>
#include <hip/hip_bf16.h>

typedef float v2f __attribute__((ext_vector_type(2)));
typedef float v8f __attribute__((ext_vector_type(8)));

#define D 128

// ---------------- zero fill ----------------
__global__ void gs_zero_kernel(float* __restrict__ p, long long n) {
    long long i = (long long)blockIdx.x * blockDim.x + threadIdx.x;
    if (i < n) p[i] = 0.0f;
}

// ---------------- degree count ----------------
__global__ void gs_count_kernel(const int* __restrict__ dst, float* __restrict__ cnt, int E) {
    int e = blockIdx.x * blockDim.x + threadIdx.x;
    if (e < E) atomicAdd(&cnt[dst[e]], 1.0f);
}

// ---------------- inv = 1/max(cnt,1) ----------------
__global__ void gs_inv_kernel(const float* __restrict__ cnt, float* __restrict__ inv, int N) {
    int i = blockIdx.x * blockDim.x + threadIdx.x;
    if (i < N) inv[i] = 1.0f / fmaxf(cnt[i], 1.0f);
}

// ---------------- edge scatter: agg[dst] += x[src] ----------------
// 32 threads per edge, 4 floats per thread (float4 gather, 4 f32 atomics)
__global__ void gs_scatter_kernel(const float* __restrict__ x,
                                  const int* __restrict__ src,
                                  const int* __restrict__ dst,
                                  float* __restrict__ agg, int E) {
    int gid = blockIdx.x * blockDim.x + threadIdx.x;
    int e = gid >> 5;
    if (e >= E) return;
    int f = (gid & 31) << 2;
    int s = src[e];
    int d = dst[e];
    const float4 v = *(const float4*)(x + (size_t)s * D + f);
    float* p = agg + (size_t)d * D + f;
    atomicAdd(p + 0, v.x);
    atomicAdd(p + 1, v.y);
    atomicAdd(p + 2, v.z);
    atomicAdd(p + 3, v.w);
}

// ---------------- fused SAGE GEMM via V_WMMA_F32_16X16X4_F32 ----------------
// out[n][o] = relu?( (agg[n]*inv[n]) . Wl[o] + x[n] . Wr[o] + bias[o] )
// One wave -> 16x64 output strip (4 WMMA o-tiles), K looped in steps of 4.
__global__ __launch_bounds__(128) void gs_sage_gemm_kernel(
    const float* __restrict__ agg, const float* __restrict__ inv,
    const float* __restrict__ xin, const float* __restrict__ Wl,
    const float* __restrict__ Wr,  const float* __restrict__ bias,
    float* __restrict__ out, int M, int do_relu)
{
    const int waveId  = (blockIdx.x * blockDim.x + threadIdx.x) >> 5;
    const int nStrips = (M / 16) * 2;           // two 64-col strips per 16-row tile
    if (waveId >= nStrips) return;              // wave-uniform exit: EXEC stays all-ones

    const int mt   = waveId >> 1;
    const int obase = (waveId & 1) * 64;
    const int lane = threadIdx.x & 31;
    const int half = lane >> 4;                 // lanes 16-31 hold the "second half"
    const int l16  = lane & 15;

    // A-fragment row for this lane (ISA: 32-bit A 16x4, M = lane&15)
    const int rowA = mt * 16 + l16;
    const float scale = inv[rowA];
    const float* aggRow = agg + (size_t)rowA * D;
    const float* xRow   = xin + (size_t)rowA * D;

    v8f acc[4] = {};                            // 4 o-tiles of 16x16 f32 C/D (8 VGPRs each)

    for (int k = 0; k < D; k += 4) {
        const int kb = k + 2 * half;            // VGPR0/1 hold K=kb,kb+1 (lanes 0-15: k,k+1; 16-31: k+2,k+3)
        v2f aA = *(const v2f*)(aggRow + kb);
        aA *= scale;                            // fold mean-divide into the A fragment
        const v2f aX = *(const v2f*)(xRow + kb);
        #pragma unroll
        for (int t = 0; t < 4; ++t) {
            const int o = obase + t * 16 + l16; // B 4x16: column = lane&15, rows kb,kb+1
            const v2f bL = *(const v2f*)(Wl + (size_t)o * D + kb); // B[k][o] = Wl[o][k]
            const v2f bR = *(const v2f*)(Wr + (size_t)o * D + kb);
            acc[t] = __builtin_amdgcn_wmma_f32_16x16x4_f32(
                false, aA, false, bL, (short)0, acc[t], false, false);
            acc[t] = __builtin_amdgcn_wmma_f32_16x16x4_f32(
                false, aX, false, bR, (short)0, acc[t], false, false);
        }
    }

    // Epilogue: C/D layout -> row = mt*16 + j + 8*half, col = obase + t*16 + l16
    #pragma unroll
    for (int t = 0; t < 4; ++t) {
        const int o = obase + t * 16 + l16;
        const float bv = bias[o];
        #pragma unroll
        for (int j = 0; j < 8; ++j) {
            const int row = mt * 16 + j + 8 * half;
            float v = acc[t][j] + bv;
            if (do_relu) v = fmaxf(v, 0.0f);
            out[(size_t)row * D + o] = v;
        }
    }
}

extern "C" void kernel_launch(void* const* d_in, const int* in_sizes, int n_in,
                              void* d_out, int out_size, void* d_ws, size_t ws_size,
                              hipStream_t stream) {
    const float* x   = (const float*)d_in[0];
    const int*   ei  = (const int*)d_in[1];
    const float* W1l = (const float*)d_in[2];
    const float* W1r = (const float*)d_in[3];
    const float* b1  = (const float*)d_in[4];
    const float* W2l = (const float*)d_in[5];
    const float* W2r = (const float*)d_in[6];
    const float* b2  = (const float*)d_in[7];

    const int N = in_sizes[0] / D;
    const int E = in_sizes[1] / 2;
    const int* src = ei;
    const int* dst = ei + E;

    // workspace layout (floats): agg[N*D] | cnt[N] | inv[N] | h[N*D]
    float* agg = (float*)d_ws;
    float* cnt = agg + (size_t)N * D;
    float* inv = cnt + N;
    float* h   = inv + N;

    float* out = (float*)d_out;

    const long long aggN = (long long)N * D;
    const int ZB = 256;
    dim3 blk(256);

    // degree counts (shared by both layers)
    gs_zero_kernel<<<dim3((unsigned)((N + ZB - 1) / ZB)), blk, 0, stream>>>(cnt, N);
    gs_count_kernel<<<dim3((unsigned)((E + ZB - 1) / ZB)), blk, 0, stream>>>(dst, cnt, E);
    gs_inv_kernel<<<dim3((unsigned)((N + ZB - 1) / ZB)), blk, 0, stream>>>(cnt, inv, N);

    // ---- layer 1 ----
    gs_zero_kernel<<<dim3((unsigned)((aggN + ZB - 1) / ZB)), blk, 0, stream>>>(agg, aggN);
    {
        long long th = (long long)E * 32;
        gs_scatter_kernel<<<dim3((unsigned)((th + ZB - 1) / ZB)), blk, 0, stream>>>(x, src, dst, agg, E);
    }
    {
        int strips = (N / 16) * 2;
        int blocks = (strips * 32 + 127) / 128;
        gs_sage_gemm_kernel<<<dim3((unsigned)blocks), dim3(128), 0, stream>>>(
            agg, inv, x, W1l, W1r, b1, h, N, 1);
    }

    // ---- layer 2 ----
    gs_zero_kernel<<<dim3((unsigned)((aggN + ZB - 1) / ZB)), blk, 0, stream>>>(agg, aggN);
    {
        long long th = (long long)E * 32;
        gs_scatter_kernel<<<dim3((unsigned)((th + ZB - 1) / ZB)), blk, 0, stream>>>(h, src, dst, agg, E);
    }
    {
        int strips = (N / 16) * 2;
        int blocks = (strips * 32 + 127) / 128;
        gs_sage_gemm_kernel<<<dim3((unsigned)blocks), dim3(128), 0, stream>>>(
            agg, inv, h, W2l, W2r, b2, out, N, 0);
    }
}